// FairEBMLayer_48885317763811
// MI455X (gfx1250) — compile-verified
//
#include <hip/hip_runtime.h>
#include <stdint.h>

#define NUM_BINS 32
#define FEATS 256
#define EDGES 33
#define PAIRS 32
#define BLOCK 512
#define WAVES (BLOCK / 32)
#define ROWS_PER_BLOCK (WAVES * 16)

typedef float v2f __attribute__((ext_vector_type(2)));
typedef float v8f __attribute__((ext_vector_type(8)));
typedef unsigned int u32x4 __attribute__((ext_vector_type(4)));
typedef int i32x4 __attribute__((ext_vector_type(4)));
typedef int i32x8 __attribute__((ext_vector_type(8)));

#if defined(__has_builtin)
#if __has_builtin(__builtin_amdgcn_tensor_load_to_lds) && \
    __has_builtin(__builtin_amdgcn_s_wait_tensorcnt)
#define HAVE_TDM 1
#endif
#if __has_builtin(__builtin_amdgcn_global_load_async_to_lds_b128) && \
    __has_builtin(__builtin_amdgcn_s_wait_asynccnt)
#define HAVE_ASYNC 1
#endif
#endif

// Compile-time probes (stderr-visible, non-fatal): which staging path built?
#if defined(HAVE_TDM)
#pragma message("FEBM staging path: TDM tensor_load_to_lds")
#elif defined(HAVE_ASYNC)
#pragma message("FEBM staging path: global_load_async_to_lds_b128")
#else
#pragma message("FEBM staging path: plain load/ds_store fallback")
#endif

// Generic->LDS(AS3) / generic->global(AS1) pointer casts (low-32 of the LDS
// generic address is the AS3 offset; aperture base has zero low bits).
#define TO_LDS(p) ((__attribute__((address_space(3))) void*)(uintptr_t)(p))
#define TO_GLB(p) ((const __attribute__((address_space(1))) void*)(uintptr_t)(p))

#if defined(HAVE_TDM)
// 1-D TDM copy: tensor = one row of nelem 4-byte elements, tile == tensor.
// D# layout per CDNA5 ISA (count=1, type=2=image,
// data_size=2 (4B), tile_dim1=1, groups 2/3 zero => dims 2..4 unused).
__device__ __forceinline__ void tdm_load_1d(const void* gsrc,
                                            unsigned lds_byte_off,
                                            unsigned nelem /* < 65536 */) {
  const unsigned long long ga = (unsigned long long)(uintptr_t)gsrc;
  u32x4 g0;
  g0[0] = 1u;                                                // count=1, user D#
  g0[1] = lds_byte_off;                                      // lds_addr
  g0[2] = (unsigned)(ga & 0xffffffffu);                      // global_addr lo
  g0[3] = (unsigned)((ga >> 32) & 0x01ffffffu) | (2u << 30); // ga hi | type=2
  i32x8 g1;
  g1[0] = (int)(2u << 16);                    // data_size = 4 bytes
  g1[1] = (int)((nelem & 0xffffu) << 16);     // tensor_dim0[15:0] @ bits 63:48
  g1[2] = (int)((nelem >> 16) | (1u << 16));  // tensor_dim0 hi | tensor_dim1=1
  g1[3] = (int)((nelem & 0xffffu) << 16);     // tile_dim0 @ bits 127:112
  g1[4] = 1;                                  // tile_dim1 = 1, tile_dim2 = 0
  g1[5] = (int)nelem;                         // tensor_dim0_stride lo32
  g1[6] = (int)((nelem & 0xffffu) << 16);     // stride0 hi=0 | stride1 lo16
  g1[7] = (int)(nelem >> 16);                 // stride1 hi
  i32x4 z4 = {0, 0, 0, 0};
#if __clang_major__ >= 23
  i32x8 z8 = {0, 0, 0, 0, 0, 0, 0, 0};
  __builtin_amdgcn_tensor_load_to_lds(g0, g1, z4, z4, z8, 0);
#else
  __builtin_amdgcn_tensor_load_to_lds(g0, g1, z4, z4, 0);
#endif
}
#endif

// Largest k in [0,31] with e[k] <= v (5-step search over 33 monotone edges);
// equivalent to the reference's clip(sum(v >= e) - 1, 0, 31).
__device__ __forceinline__ int bsearch32(const float* __restrict__ e, float v) {
  int lo = (v >= e[16]) ? 16 : 0;
  lo += (v >= e[lo + 8]) ? 8 : 0;
  lo += (v >= e[lo + 4]) ? 4 : 0;
  lo += (v >= e[lo + 2]) ? 2 : 0;
  lo += (v >= e[lo + 1]) ? 1 : 0;
  return lo;
}

__global__ void febm_init(int* __restrict__ hist_g) {
  if (threadIdx.x < 64) hist_g[threadIdx.x] = 0;
}

__global__ __launch_bounds__(BLOCK, 1) void febm_main(
    const float* __restrict__ x, const float* __restrict__ W,
    const float* __restrict__ IW, const float* __restrict__ intercept,
    const float* __restrict__ bins, const int* __restrict__ pair_i,
    const int* __restrict__ pair_j, float* __restrict__ out,
    int* __restrict__ hist_g) {
  __shared__ float bins_s[FEATS * EDGES];              // 33,792 B
  __shared__ float W_s[FEATS * NUM_BINS];              // 32,768 B
  __shared__ float IW_s[PAIRS * NUM_BINS * NUM_BINS];  // 131,072 B
  __shared__ unsigned char idx_s[WAVES * 16 * 64];     // 16,384 B
  __shared__ float red_s[WAVES * 16];                  //  1,024 B
  __shared__ int hist_s[64];
  __shared__ int pi_s[PAIRS], pj_s[PAIRS];

  const int tid = threadIdx.x;

  // ---- Stage hot tables into LDS via the CDNA5 async/tensor path ----
#if defined(HAVE_TDM)
  if (tid == 0) {  // one wave issues 3 in-order TDM descriptors (EXEC ignored)
    tdm_load_1d(bins, (unsigned)(uintptr_t)&bins_s[0], FEATS * EDGES);
    tdm_load_1d(W, (unsigned)(uintptr_t)&W_s[0], FEATS * NUM_BINS);
    tdm_load_1d(IW, (unsigned)(uintptr_t)&IW_s[0], PAIRS * NUM_BINS * NUM_BINS);
  }
#elif defined(HAVE_ASYNC)
  {
    const float4* b4 = (const float4*)bins;
    const float4* w4 = (const float4*)W;
    const float4* i4 = (const float4*)IW;
    float4* bs4 = (float4*)bins_s;
    float4* ws4 = (float4*)W_s;
    float4* is4 = (float4*)IW_s;
    for (int i = tid; i < (FEATS * EDGES) / 4; i += BLOCK)
      __builtin_amdgcn_global_load_async_to_lds_b128(TO_GLB(b4 + i),
                                                     TO_LDS(bs4 + i), 0, 0);
    for (int i = tid; i < (FEATS * NUM_BINS) / 4; i += BLOCK)
      __builtin_amdgcn_global_load_async_to_lds_b128(TO_GLB(w4 + i),
                                                     TO_LDS(ws4 + i), 0, 0);
    for (int i = tid; i < (PAIRS * NUM_BINS * NUM_BINS) / 4; i += BLOCK)
      __builtin_amdgcn_global_load_async_to_lds_b128(TO_GLB(i4 + i),
                                                     TO_LDS(is4 + i), 0, 0);
  }
#else
  for (int i = tid; i < FEATS * EDGES; i += BLOCK) bins_s[i] = bins[i];
  for (int i = tid; i < FEATS * NUM_BINS; i += BLOCK) W_s[i] = W[i];
  for (int i = tid; i < PAIRS * NUM_BINS * NUM_BINS; i += BLOCK) IW_s[i] = IW[i];
#endif
  if (tid < 64) hist_s[tid] = 0;
  if (tid < PAIRS) {
    pi_s[tid] = pair_i[tid];
    pj_s[tid] = pair_j[tid];
  }
#if defined(HAVE_TDM)
  __builtin_amdgcn_s_wait_tensorcnt(0);  // wave 0 drains TENSORcnt; others NOP
#elif defined(HAVE_ASYNC)
  __builtin_amdgcn_s_wait_asynccnt(0);
#endif
  __syncthreads();

  const int wave = tid >> 5;
  const int lane = tid & 31;
  const int row16 = lane & 15;  // wave handles 16 rows; lane -> row
  const int kh = lane >> 4;     // 0: K=0,1   1: K=2,3  (WMMA f32 A layout)
  const int r = blockIdx.x * ROWS_PER_BLOCK + wave * 16 + row16;
  const float* __restrict__ xr = x + (size_t)r * FEATS;
  const int wslot = (wave * 16 + row16) * 64;

  v8f acc = {};  // 16x16 f32 accumulator; column 0 = row sums
  v2f bones;     // B = ones -> D[m][n] = sum_k A[m][k] + C[m][n]
  bones.x = 1.0f;
  bones.y = 1.0f;

  for (int f0 = 0; f0 < FEATS; f0 += 4) {
    const int fa = f0 + 2 * kh;
    __builtin_prefetch(xr + fa + 96, 0, 3);       // global_prefetch_b8 ahead
    const float2 xv = *(const float2*)(xr + fa);  // fa even -> 8B aligned
    const int ia = bsearch32(&bins_s[fa * EDGES], xv.x);
    const int ib = bsearch32(&bins_s[(fa + 1) * EDGES], xv.y);
    const float m0 = W_s[fa * NUM_BINS + ia];
    const float m1 = W_s[(fa + 1) * NUM_BINS + ib];

    if (f0 < 64) {  // interaction features are 0..63
      idx_s[wslot + fa] = (unsigned char)ia;
      idx_s[wslot + fa + 1] = (unsigned char)ib;
    }
    // Sensitive features: f==0 (f0==0,kh==0,elem0), f==5 (f0==4,kh==0,elem1)
    if (f0 == 0 && kh == 0) atomicAdd(&hist_s[ia], 1);
    if (f0 == 4 && kh == 0) atomicAdd(&hist_s[32 + ib], 1);

    v2f a;
    a.x = m0;  // K even
    a.y = m1;  // K odd
    // D = A(16x4) * ones(4x16) + C : accumulates 4 features per issue.
    acc = __builtin_amdgcn_wmma_f32_16x16x4_f32(false, a, false, bones,
                                                (short)0, acc, false, false);
  }

  // Column 0 of D: lane 0 holds rows 0..7 in acc[0..7], lane 16 rows 8..15.
  if (lane == 0)
    for (int m = 0; m < 8; ++m) red_s[wave * 16 + m] = acc[m];
  if (lane == 16)
    for (int m = 0; m < 8; ++m) red_s[wave * 16 + 8 + m] = acc[m];
  __syncthreads();

  // Pairwise interaction gathers + final prediction (lanes 0..15 per wave).
  if (lane < 16) {
    float s = red_s[wave * 16 + row16] + intercept[0];
    const unsigned char* __restrict__ myidx = &idx_s[wslot];
    for (int p = 0; p < PAIRS; ++p) {
      const int ii = myidx[pi_s[p]];
      const int jj = myidx[pj_s[p]];
      s += IW_s[p * (NUM_BINS * NUM_BINS) + ii * NUM_BINS + jj];
    }
    out[r] = s;
  }

  __syncthreads();
  if (tid < 64) atomicAdd(&hist_g[tid], hist_s[tid]);
}

__global__ void febm_finalize(const float* __restrict__ W,
                              const int* __restrict__ hist_g,
                              float* __restrict__ out_f, int Brows) {
  if (threadIdx.x == 0) {
    const int sens[2] = {0, 5};
    float fl = 0.0f;
    const float invB = 1.0f / (float)Brows;
    for (int s = 0; s < 2; ++s) {
      float s1 = 0.0f, s2 = 0.0f;
      for (int k = 0; k < NUM_BINS; ++k) {
        const float w = W[sens[s] * NUM_BINS + k];
        const float c = (float)hist_g[s * 32 + k];
        s1 += c * w;
        s2 += c * w * w;
      }
      const float mean = s1 * invB;
      fl += 0.1f * (s2 * invB - mean * mean);
    }
    *out_f = fl;
  }
}

extern "C" void kernel_launch(void* const* d_in, const int* in_sizes, int n_in,
                              void* d_out, int out_size, void* d_ws,
                              size_t ws_size, hipStream_t stream) {
  const float* x = (const float*)d_in[0];
  const float* W = (const float*)d_in[1];
  const float* IW = (const float*)d_in[2];
  const float* intercept = (const float*)d_in[3];
  const float* bins = (const float*)d_in[4];
  const int* pair_i = (const int*)d_in[5];
  const int* pair_j = (const int*)d_in[6];
  float* out = (float*)d_out;

  const int Brows = in_sizes[0] / FEATS;  // 65536
  int* hist_g = (int*)d_ws;               // 64 ints of scratch

  febm_init<<<1, 64, 0, stream>>>(hist_g);
  febm_main<<<Brows / ROWS_PER_BLOCK, BLOCK, 0, stream>>>(
      x, W, IW, intercept, bins, pair_i, pair_j, out, hist_g);
  febm_finalize<<<1, 64, 0, stream>>>(W, hist_g, out + Brows, Brows);
}